// GCN_48550310313992
// MI455X (gfx1250) — compile-verified
//
#include <hip/hip_runtime.h>
#include <math.h>

typedef __attribute__((ext_vector_type(2))) float v2f;
typedef __attribute__((ext_vector_type(8))) float v8f;

#define IN_FEATS 512
#define HIDDEN   16
#define NCLASS   7
#define H2PAD    8

// ---------------- degree / normalization ----------------

__global__ void deg_init_kernel(float* __restrict__ deg, int N) {
    int i = blockIdx.x * blockDim.x + threadIdx.x;
    if (i < N) deg[i] = 1.0f;   // self-loop contributes 1
}

__global__ void deg_edges_kernel(const long long* __restrict__ ei, float* __restrict__ deg,
                                 int E) {
    int e = blockIdx.x * blockDim.x + threadIdx.x;
    if (e < E) {
        int d = (int)ei[(size_t)E + e];   // dst row
        atomicAdd(&deg[d], 1.0f);
    }
}

__global__ void dinv_kernel(float* __restrict__ deg, int N) {
    int i = blockIdx.x * blockDim.x + threadIdx.x;
    if (i < N) deg[i] = __frsqrt_rn(deg[i]);   // deg >= 1 always
}

// ---------------- layer 1 GEMM: h1 = x @ W1 (fp32 WMMA) ----------------
// One wave per TWO 16-row tiles (32 rows): both tiles share the same B
// fragment (halves LDS traffic) and form two independent WMMA accumulation
// chains (hides accumulate latency in the matrix pipe). K = 512.

__global__ void gemm1_kernel(const float* __restrict__ x, const float* __restrict__ W1,
                             float* __restrict__ h1, int nTiles2) {
    __shared__ float sW[IN_FEATS * HIDDEN];   // 32 KB
    for (int i = threadIdx.x; i < IN_FEATS * HIDDEN / 4; i += blockDim.x)
        ((float4*)sW)[i] = ((const float4*)W1)[i];
    __syncthreads();

    int wave = threadIdx.x >> 5;
    int lane = threadIdx.x & 31;
    int t2   = blockIdx.x * 4 + wave;          // wave-uniform
    if (t2 >= nTiles2) return;

    int row0 = t2 * 32;
    int m    = lane & 15;                      // A: M index, B: N index
    int kk   = (lane >> 4) << 1;               // 0 for lanes 0-15, 2 for lanes 16-31
    const float* a0 = x + (size_t)(row0 + m) * IN_FEATS + kk;
    const float* a1 = a0 + (size_t)16 * IN_FEATS;

    v8f c0 = {}, c1 = {};
    #pragma unroll 8
    for (int k = 0; k < IN_FEATS; k += 4) {
        v2f b;
        b.x = sW[(k + kk) * HIDDEN + m];       // W1[k+kk,   n]
        b.y = sW[(k + kk + 1) * HIDDEN + m];   // W1[k+kk+1, n]
        v2f va0 = *(const v2f*)(a0 + k);
        v2f va1 = *(const v2f*)(a1 + k);
        c0 = __builtin_amdgcn_wmma_f32_16x16x4_f32(false, va0, false, b, (short)0, c0,
                                                   false, false);
        c1 = __builtin_amdgcn_wmma_f32_16x16x4_f32(false, va1, false, b, (short)0, c1,
                                                   false, false);
    }

    // C/D layout: VGPR j -> M = j (lanes 0-15) or j+8 (lanes 16-31), N = lane&15
    int rbase = row0 + ((lane >> 4) << 3);
    #pragma unroll
    for (int j = 0; j < 8; ++j)
        h1[(size_t)(rbase + j) * HIDDEN + m] = c0[j];
    #pragma unroll
    for (int j = 0; j < 8; ++j)
        h1[(size_t)(rbase + 16 + j) * HIDDEN + m] = c1[j];
}

// ---------------- layer 1 aggregation ----------------
// agg1[i,f] = b1[f] + h1[i,f]*dinv[i]^2   (self loop folded in)

__global__ void agg1_init_kernel(const float* __restrict__ h1, const float* __restrict__ b1,
                                 const float* __restrict__ dinv, float* __restrict__ agg1,
                                 int N) {
    int t = blockIdx.x * blockDim.x + threadIdx.x;
    if (t < N * HIDDEN) {
        int i = t >> 4, f = t & 15;
        float w = dinv[i];
        agg1[t] = b1[f] + h1[t] * w * w;
    }
}

__global__ void agg1_edges_kernel(const long long* __restrict__ ei,
                                  const float* __restrict__ h1,
                                  const float* __restrict__ dinv,
                                  float* __restrict__ agg1, int E) {
    int idx = blockIdx.x * blockDim.x + threadIdx.x;
    int e = idx >> 4;
    if (e >= E) return;
    int f = idx & 15;
    int s = (int)ei[e];
    int d = (int)ei[(size_t)E + e];
    float w = dinv[s] * dinv[d];
    atomicAdd(&agg1[(size_t)d * HIDDEN + f], h1[(size_t)s * HIDDEN + f] * w);
}

// ---------------- layer 2 GEMM: h2 = relu(agg1) @ W2 (fp32 WMMA, N padded to 16) ----

__global__ void gemm2_kernel(const float* __restrict__ agg1, const float* __restrict__ W2,
                             float* __restrict__ h2, int nTiles2) {
    __shared__ float sW[16 * 16];
    if (threadIdx.x < 256) {
        int k = threadIdx.x >> 4, n = threadIdx.x & 15;
        sW[threadIdx.x] = (n < NCLASS) ? W2[k * NCLASS + n] : 0.0f;
    }
    __syncthreads();

    int wave = threadIdx.x >> 5;
    int lane = threadIdx.x & 31;
    int t2   = blockIdx.x * 4 + wave;
    if (t2 >= nTiles2) return;

    int row0 = t2 * 32;
    int m    = lane & 15;
    int kk   = (lane >> 4) << 1;
    const float* a0 = agg1 + (size_t)(row0 + m) * HIDDEN + kk;
    const float* a1 = a0 + (size_t)16 * HIDDEN;

    v8f c0 = {}, c1 = {};
    #pragma unroll
    for (int k = 0; k < HIDDEN; k += 4) {
        v2f b;
        b.x = sW[(k + kk) * 16 + m];
        b.y = sW[(k + kk + 1) * 16 + m];
        v2f va0, va1;
        va0.x = fmaxf(a0[k], 0.0f);            // relu fused into A load
        va0.y = fmaxf(a0[k + 1], 0.0f);
        va1.x = fmaxf(a1[k], 0.0f);
        va1.y = fmaxf(a1[k + 1], 0.0f);
        c0 = __builtin_amdgcn_wmma_f32_16x16x4_f32(false, va0, false, b, (short)0, c0,
                                                   false, false);
        c1 = __builtin_amdgcn_wmma_f32_16x16x4_f32(false, va1, false, b, (short)0, c1,
                                                   false, false);
    }

    int rbase = row0 + ((lane >> 4) << 3);
    if (m < H2PAD) {                            // cols 0..6 valid, col 7 == 0
        #pragma unroll
        for (int j = 0; j < 8; ++j)
            h2[(size_t)(rbase + j) * H2PAD + m] = c0[j];
        #pragma unroll
        for (int j = 0; j < 8; ++j)
            h2[(size_t)(rbase + 16 + j) * H2PAD + m] = c1[j];
    }
}

// ---------------- layer 2 aggregation ----------------

__global__ void agg2_init_kernel(const float* __restrict__ h2, const float* __restrict__ b2,
                                 const float* __restrict__ dinv, float* __restrict__ agg2,
                                 int N) {
    int t = blockIdx.x * blockDim.x + threadIdx.x;
    if (t < N * H2PAD) {
        int i = t >> 3, c = t & 7;
        float w = dinv[i];
        agg2[t] = ((c < NCLASS) ? b2[c] : 0.0f) + h2[t] * w * w;
    }
}

__global__ void agg2_edges_kernel(const long long* __restrict__ ei,
                                  const float* __restrict__ h2,
                                  const float* __restrict__ dinv,
                                  float* __restrict__ agg2, int E) {
    int idx = blockIdx.x * blockDim.x + threadIdx.x;
    int e = idx >> 3;
    if (e >= E) return;
    int c = idx & 7;
    if (c >= NCLASS) return;
    int s = (int)ei[e];
    int d = (int)ei[(size_t)E + e];
    float w = dinv[s] * dinv[d];
    atomicAdd(&agg2[(size_t)d * H2PAD + c], h2[(size_t)s * H2PAD + c] * w);
}

// ---------------- row-wise log_softmax over 7 classes ----------------

__global__ void logsoftmax_kernel(const float* __restrict__ agg2, float* __restrict__ out,
                                  int N) {
    int i = blockIdx.x * blockDim.x + threadIdx.x;
    if (i >= N) return;
    float v[NCLASS];
    float mx = -INFINITY;
    #pragma unroll
    for (int c = 0; c < NCLASS; ++c) {
        v[c] = agg2[(size_t)i * H2PAD + c];
        mx = fmaxf(mx, v[c]);
    }
    float s = 0.0f;
    #pragma unroll
    for (int c = 0; c < NCLASS; ++c) s += __expf(v[c] - mx);
    float lse = mx + __logf(s);
    #pragma unroll
    for (int c = 0; c < NCLASS; ++c) out[(size_t)i * NCLASS + c] = v[c] - lse;
}

// ---------------- host launcher ----------------

extern "C" void kernel_launch(void* const* d_in, const int* in_sizes, int n_in,
                              void* d_out, int out_size, void* d_ws, size_t ws_size,
                              hipStream_t stream) {
    const float*     x  = (const float*)d_in[0];
    const long long* ei = (const long long*)d_in[1];   // int64 [2, E]
    const float*     W1 = (const float*)d_in[2];
    const float*     b1 = (const float*)d_in[3];
    const float*     W2 = (const float*)d_in[4];
    const float*     b2 = (const float*)d_in[5];
    float* out = (float*)d_out;

    const int N = in_sizes[0] / IN_FEATS;   // 100000 (multiple of 32)
    const int E = in_sizes[1] / 2;          // 1600000
    const int nTiles2 = N / 32;             // 3125 double-tiles

    // workspace layout (floats)
    float* dinv = (float*)d_ws;                     // N
    float* h1   = dinv + ((N + 255) & ~255);        // N*16
    float* agg1 = h1   + (size_t)N * HIDDEN;        // N*16
    float* h2   = agg1 + (size_t)N * HIDDEN;        // N*8
    float* agg2 = h2   + (size_t)N * H2PAD;         // N*8

    const int T = 256;

    deg_init_kernel<<<(N + T - 1) / T, T, 0, stream>>>(dinv, N);
    deg_edges_kernel<<<(E + T - 1) / T, T, 0, stream>>>(ei, dinv, E);
    dinv_kernel<<<(N + T - 1) / T, T, 0, stream>>>(dinv, N);

    gemm1_kernel<<<(nTiles2 + 3) / 4, 128, 0, stream>>>(x, W1, h1, nTiles2);

    agg1_init_kernel<<<((size_t)N * HIDDEN + T - 1) / T, T, 0, stream>>>(h1, b1, dinv, agg1, N);
    agg1_edges_kernel<<<((size_t)E * 16 + T - 1) / T, T, 0, stream>>>(ei, h1, dinv, agg1, E);

    gemm2_kernel<<<(nTiles2 + 3) / 4, 128, 0, stream>>>(agg1, W2, h2, nTiles2);

    agg2_init_kernel<<<((size_t)N * H2PAD + T - 1) / T, T, 0, stream>>>(h2, b2, dinv, agg2, N);
    agg2_edges_kernel<<<((size_t)E * 8 + T - 1) / T, T, 0, stream>>>(ei, h2, dinv, agg2, E);

    logsoftmax_kernel<<<(N + T - 1) / T, T, 0, stream>>>(agg2, out, N);
}